// EllipsoidKernelLoss_29197187678597
// MI455X (gfx1250) — compile-verified
//
#include <hip/hip_runtime.h>
#include <hip/hip_bf16.h>

// ---------------------------------------------------------------------------
// EllipsoidKernelLoss for MI455X (gfx1250, wave32).
//   grid:  B*N = 320 blocks, 256 threads (8 wave32) each
//   inner: thread owns column j = tid; loops 256 rows; loop-invariant dx terms
//   reduce: V_WMMA_F32_16X16X4_F32 (A = partials, B = ones) -> 32-lane sum,
//           then LDS combine across the 8 waves, deterministic final pass.
// ---------------------------------------------------------------------------

typedef float v2f __attribute__((ext_vector_type(2)));
typedef float v8f __attribute__((ext_vector_type(8)));

#define HH 256
#define WW 256
#define BB 16
#define NN 20
#define NBLK (BB * NN)

__device__ __forceinline__ float wave_reduce_wmma_f32(float x) {
    // A: 16x4 F32. Lanes 0-15 supply A[m][0],A[m][1]; lanes 16-31 supply
    // A[m][2],A[m][3]. Put the partial in the first slot, 0 in the second.
    v2f a; a.x = x; a.y = 0.0f;
    // B: 4x16 all ones -> D[m][n] = A[m][0]+A[m][1]+A[m][2]+A[m][3]
    //                             = x[m] + x[m+16]   (same for every column n)
    v2f b; b.x = 1.0f; b.y = 1.0f;
    v8f c = {};
    v8f d = __builtin_amdgcn_wmma_f32_16x16x4_f32(
        /*neg_a=*/false, a, /*neg_b=*/false, b,
        /*c_mod=*/(short)0, c, /*reuse_a=*/false, /*reuse_b=*/false);
    // Lane L<16 holds rows 0..7 of some column; lane L>=16 holds rows 8..15.
    float s = d[0] + d[1] + d[2] + d[3] + d[4] + d[5] + d[6] + d[7];
    // rows 0..7 partial + rows 8..15 partial -> full 32-lane sum in all lanes
    s += __shfl_xor(s, 16, 32);
    return s;
}

__global__ void __launch_bounds__(256)
ellipsoid_loss_phase1(const float* __restrict__ pred,      // [B,N,2]
                      const float* __restrict__ target,    // [B,N,2]
                      const float* __restrict__ vis,       // [B,N]
                      const int*   __restrict__ pairs,     // [B,2]
                      float* __restrict__ partial)         // [B*N]
{
    const int bn = blockIdx.x;          // 0..319
    const int b  = bn / NN;
    const int n  = bn % NN;
    const int tid  = threadIdx.x;       // 0..255  == pixel column j
    const int lane = tid & 31;
    const int wid  = tid >> 5;

    // ---- per-(b,n) scalars (uniform; broadcast loads) ----
    const float tx = target[bn * 2 + 0];
    const float ty = target[bn * 2 + 1];
    const float px = pred[bn * 2 + 0];
    const float py = pred[bn * 2 + 1];
    const float v  = vis[bn];

    const int i0 = pairs[b * 2 + 0];
    const int i1 = pairs[b * 2 + 1];
    const float bx = target[(b * NN + i0) * 2 + 0];
    const float by = target[(b * NN + i0) * 2 + 1];
    const float hx = target[(b * NN + i1) * 2 + 0];
    const float hy = target[(b * NN + i1) * 2 + 1];
    const float vb = vis[b * NN + i0];
    const float vh = vis[b * NN + i1];

    const float ddx = hx - bx, ddy = hy - by;
    const float nrm = sqrtf(ddx * ddx + ddy * ddy);
    const float ang = (vb > 0.5f && vh > 0.5f && nrm > 1e-6f)
                        ? atan2f(ddy, ddx) : 0.0f;
    const float ca = cosf(ang);
    const float sa = sinf(ang);

    // ---- loop-invariant column terms ----
    const float inv255 = 1.0f / 255.0f;           // linspace(0,1,256) step
    const float gx   = (float)tid * inv255;
    const float dx   = gx - tx;
    const float pdx  = gx - px;
    const float dxca = dx * ca;
    const float dxsa = dx * sa;
    const float dx2  = dx * dx;
    const float pdx2 = pdx * pdx;

    const float c1 = 1.0f / (2.0f * 0.15f * 0.15f);   // 1/(2*(SIG_A/SHARP)^2)
    const float c2 = 1.0f / (2.0f * 0.05f * 0.05f);   // 1/(2*(SIG_B/SHARP)^2)
    const float R2 = 0.3f * 0.3f;

    float sw = 0.0f, swd = 0.0f;
#pragma unroll 4
    for (int i = 0; i < HH; ++i) {
        const float gy = (float)i * inv255;
        const float dy = gy - ty;
        const float d2 = dx2 + dy * dy;                 // dist^2
        const float xr = dxca + dy * sa;
        const float yr = dy * ca - dxsa;
        const float ed = xr * xr * c1 + yr * yr * c2;   // ed/2 of reference
        const float w  = (d2 <= R2) ? __expf(-ed) : 0.0f;
        const float pdy = gy - py;
        const float pd  = sqrtf(pdx2 + pdy * pdy);
        sw  += w;
        swd += w * pd;
    }

    // ---- wave32 reduction on the matrix pipe (EXEC all ones here) ----
    const float wsw  = wave_reduce_wmma_f32(sw);
    const float wswd = wave_reduce_wmma_f32(swd);

    __shared__ float s_sw[8];
    __shared__ float s_swd[8];
    if (lane == 0) { s_sw[wid] = wsw; s_swd[wid] = wswd; }
    __syncthreads();

    if (tid == 0) {
        float SW = 0.0f, SWD = 0.0f;
#pragma unroll
        for (int k = 0; k < 8; ++k) { SW += s_sw[k]; SWD += s_swd[k]; }
        const float per = SWD / (SW + 1e-8f);
        partial[bn] = (v >= 0.5f) ? per : 0.0f;
    }
}

__global__ void ellipsoid_loss_phase2(const float* __restrict__ partial,
                                      float* __restrict__ out)
{
    if (blockIdx.x == 0 && threadIdx.x == 0) {
        // fixed sequential order -> bitwise-deterministic result
        float t = 0.0f;
        for (int k = 0; k < NBLK; ++k) t += partial[k];
        out[0] = t / ((float)NBLK + 1e-8f);
    }
}

extern "C" void kernel_launch(void* const* d_in, const int* in_sizes, int n_in,
                              void* d_out, int out_size, void* d_ws, size_t ws_size,
                              hipStream_t stream) {
    const float* pred   = (const float*)d_in[0];   // [16,20,2]
    const float* target = (const float*)d_in[1];   // [16,20,2]
    const float* vis    = (const float*)d_in[2];   // [16,20]
    const int*   pairs  = (const int*)d_in[3];     // [16,2]
    float* out = (float*)d_out;
    float* ws  = (float*)d_ws;                     // 320 floats of scratch

    ellipsoid_loss_phase1<<<NBLK, 256, 0, stream>>>(pred, target, vis, pairs, ws);
    ellipsoid_loss_phase2<<<1, 32, 0, stream>>>(ws, out);
}